// InteractionNetwork_27023934227051
// MI455X (gfx1250) — compile-verified
//
#include <hip/hip_runtime.h>

// InteractionNetwork on MI455X (gfx1250), wave32 + f32 WMMA.
//
// Key algebraic rewrite: agg = (sum_j A[i,j] * relu(xi_p[i]+xj_p[j]+eb1)) @ ew2
//                              + (sum_j A[i,j]) * eb2
// so the O(N^2) per-edge GEMM of the reference collapses to per-node GEMMs,
// and the only O(N^2) work left is elementwise relu+fma (VALU), LDS-staged.
//
// All GEMMs are [16-row tiles, K=128] x [128,128] using chained
// V_WMMA_F32_16X16X4_F32 (32 steps per tile), per the CDNA5 ISA VGPR layouts:
//   A 16x4 f32 : lane half h=lane>>4, m=lane&15 -> a = {X[m][k+2h], X[m][k+2h+1]}
//   B 4x16 f32 : b = {W[k+2h][c0+m], W[k+2h+1][c0+m]}
//   C 16x16 f32: vgpr r, lane -> row r+8h, col c0+m

typedef __attribute__((ext_vector_type(2))) float v2f;
typedef __attribute__((ext_vector_type(8))) float v8f;

#define BN 1024        // B*N rows
#define NN 256         // nodes per batch
#define DD 128         // feature dim
#define LDT 132        // padded LDS row stride (132*4B = 528B, 16B aligned, spreads banks)

__device__ __forceinline__ v8f wmma_gemm_k128(const float* __restrict__ Alds, int lda,
                                              const float* __restrict__ Bg, int ldb,
                                              v8f acc, int half, int m) {
  // acc += Alds[16 x 128] @ Bg[128 x 16]   (Bg already offset to column base)
#pragma unroll
  for (int kk = 0; kk < DD; kk += 4) {
    const int ka = kk + (half << 1);
    v2f a, b;
    a.x = Alds[m * lda + ka];
    a.y = Alds[m * lda + ka + 1];
    b.x = Bg[ka * ldb + m];
    b.y = Bg[(ka + 1) * ldb + m];
    acc = __builtin_amdgcn_wmma_f32_16x16x4_f32(false, a, false, b, (short)0, acc,
                                                false, false);
  }
  return acc;
}

// ---------------------------------------------------------------------------
// Kernel 1: xi_p = x @ ew1[l,:D] + eb1  ;  xj_p = x @ ew1[l,D:]
// grid = 64 row-tiles of 16, block = 256 (8 waves). Wave handles 2 col tiles.
// ---------------------------------------------------------------------------
__global__ __launch_bounds__(256) void in_proj_kernel(
    const float* __restrict__ x, const float* __restrict__ ew1l,
    const float* __restrict__ eb1l, float* __restrict__ xi_p,
    float* __restrict__ xj_p) {
  __shared__ float xlds[16 * LDT];
  const int tid = threadIdx.x;
  const int row0 = blockIdx.x * 16;

  // stage 16x128 X tile into LDS (float4, coalesced)
  for (int idx = tid; idx < 512; idx += 256) {
    const int r = idx >> 5, c = (idx & 31) << 2;
    *(float4*)&xlds[r * LDT + c] = *(const float4*)&x[(row0 + r) * DD + c];
  }
  __syncthreads();

  const int wave = tid >> 5, lane = tid & 31;
  const int half = lane >> 4, m = lane & 15;

#pragma unroll
  for (int tt = 0; tt < 2; ++tt) {
    const int t = wave + tt * 8;          // 0..15 col tiles across [xi_p | xj_p]
    const int c0 = (t & 7) * 16;
    const float* Wg = ew1l + (t >= 8 ? DD * DD : 0) + c0;
    const float bias = (t < 8) ? eb1l[c0 + m] : 0.0f;  // fold eb1 into xi_p
    v8f acc;
#pragma unroll
    for (int r = 0; r < 8; ++r) acc[r] = bias;
    acc = wmma_gemm_k128(xlds, LDT, Wg, DD, acc, half, m);
    float* out = (t < 8) ? xi_p : xj_p;
#pragma unroll
    for (int r = 0; r < 8; ++r)
      out[(row0 + r + 8 * half) * DD + c0 + m] = acc[r];
  }
}

// ---------------------------------------------------------------------------
// Kernel 2 (fused): per block = 16 rows of one batch.
//   phase 1 (VALU): hsum[i,d] = sum_j A[i,j]*relu(xi_p[i,d]+xj_p[j,d]), asum[i]
//   phase 2 (WMMA): agg = hsum @ ew2 + asum (x) eb2
//   phase 3 (WMMA): nh  = relu(x @ nw1a + agg @ nw1b + nb1)
//   phase 4 (WMMA): xout = x + nh @ nw2 + nb2
// ---------------------------------------------------------------------------
__global__ __launch_bounds__(256) void in_fused_kernel(
    const float* __restrict__ x, const float* __restrict__ adj,
    const float* __restrict__ xi_p, const float* __restrict__ xj_p,
    const float* __restrict__ ew2l, const float* __restrict__ eb2l,
    const float* __restrict__ nw1l, const float* __restrict__ nb1l,
    const float* __restrict__ nw2l, const float* __restrict__ nb2l,
    float* __restrict__ xout) {
  __shared__ float xrow[16 * LDT];   // x tile (MLP input + residual)
  __shared__ float xjb[32 * DD];     // xj_p chunk
  __shared__ float Ab[16 * 32];      // adjacency chunk
  __shared__ float hs[16 * LDT];     // hsum
  __shared__ float ag[16 * LDT];     // agg
  __shared__ float nhb[16 * LDT];    // node-MLP hidden
  __shared__ float asumb[16];

  const int tid = threadIdx.x;
  const int b = blockIdx.x >> 4;
  const int i0 = (blockIdx.x & 15) << 4;
  const int grow0 = b * NN + i0;

  // stage the 16 x rows
  for (int idx = tid; idx < 512; idx += 256) {
    const int r = idx >> 5, c = (idx & 31) << 2;
    *(float4*)&xrow[r * LDT + c] = *(const float4*)&x[(grow0 + r) * DD + c];
  }

  // ---- phase 1: thread owns (row i, 8 features d0..d0+7) ----
  const int i = tid >> 4;
  const int dg = tid & 15;
  const int d0 = dg << 3;

  float xi[8];
  {
    const float4 a = *(const float4*)&xi_p[(grow0 + i) * DD + d0];
    const float4 c = *(const float4*)&xi_p[(grow0 + i) * DD + d0 + 4];
    xi[0] = a.x; xi[1] = a.y; xi[2] = a.z; xi[3] = a.w;
    xi[4] = c.x; xi[5] = c.y; xi[6] = c.z; xi[7] = c.w;
  }
  float acc[8] = {0.f, 0.f, 0.f, 0.f, 0.f, 0.f, 0.f, 0.f};
  float awsum = 0.f;

  for (int jc = 0; jc < NN; jc += 32) {
    __syncthreads();  // previous chunk fully consumed (also covers xrow store)
    for (int idx = tid; idx < 1024; idx += 256) {   // 32x128 xj_p chunk
      const int r = idx >> 5, c = (idx & 31) << 2;
      *(float4*)&xjb[r * DD + c] = *(const float4*)&xj_p[(b * NN + jc + r) * DD + c];
    }
    for (int idx = tid; idx < 512; idx += 256) {    // 16x32 adjacency chunk
      const int r = idx >> 5, cj = idx & 31;
      Ab[r * 32 + cj] = adj[b * NN * NN + (i0 + r) * NN + jc + cj];
    }
    __syncthreads();
#pragma unroll 4
    for (int jj = 0; jj < 32; ++jj) {
      const float aw = Ab[i * 32 + jj];             // broadcast ds read
      const float4 xa = *(const float4*)&xjb[jj * DD + d0];
      const float4 xb = *(const float4*)&xjb[jj * DD + d0 + 4];
      const float xjv[8] = {xa.x, xa.y, xa.z, xa.w, xb.x, xb.y, xb.z, xb.w};
      awsum += aw;
#pragma unroll
      for (int q = 0; q < 8; ++q) {
        const float h = fmaxf(xi[q] + xjv[q], 0.0f);
        acc[q] = fmaf(aw, h, acc[q]);
      }
    }
  }

#pragma unroll
  for (int q = 0; q < 8; ++q) hs[i * LDT + d0 + q] = acc[q];
  if (dg == 0) asumb[i] = awsum;
  __syncthreads();

  // ---- WMMA phases: wave w owns col tile c0 = 16w ----
  const int wave = tid >> 5, lane = tid & 31;
  const int half = lane >> 4, m = lane & 15;
  const int c0 = wave * 16;

  {  // phase 2: agg = hs @ ew2 + asum * eb2
    v8f a;
    const float e2 = eb2l[c0 + m];
#pragma unroll
    for (int r = 0; r < 8; ++r) a[r] = asumb[r + 8 * half] * e2;
    a = wmma_gemm_k128(hs, LDT, ew2l + c0, DD, a, half, m);
#pragma unroll
    for (int r = 0; r < 8; ++r) ag[(r + 8 * half) * LDT + c0 + m] = a[r];
  }
  __syncthreads();

  {  // phase 3: nh = relu(x @ nw1a + agg @ nw1b + nb1)
    v8f a;
    const float bias = nb1l[c0 + m];
#pragma unroll
    for (int r = 0; r < 8; ++r) a[r] = bias;
    a = wmma_gemm_k128(xrow, LDT, nw1l + c0, DD, a, half, m);
    a = wmma_gemm_k128(ag, LDT, nw1l + DD * DD + c0, DD, a, half, m);
#pragma unroll
    for (int r = 0; r < 8; ++r)
      nhb[(r + 8 * half) * LDT + c0 + m] = fmaxf(a[r], 0.0f);
  }
  __syncthreads();

  {  // phase 4: xout = x + nh @ nw2 + nb2
    v8f a;
    const float bias = nb2l[c0 + m];
#pragma unroll
    for (int r = 0; r < 8; ++r) a[r] = bias;
    a = wmma_gemm_k128(nhb, LDT, nw2l + c0, DD, a, half, m);
#pragma unroll
    for (int r = 0; r < 8; ++r) {
      const int row = r + 8 * half;
      xout[(grow0 + row) * DD + c0 + m] = a[r] + xrow[row * LDT + c0 + m];
    }
  }
}

extern "C" void kernel_launch(void* const* d_in, const int* in_sizes, int n_in,
                              void* d_out, int out_size, void* d_ws, size_t ws_size,
                              hipStream_t stream) {
  const float* x0  = (const float*)d_in[0];
  const float* adj = (const float*)d_in[1];
  const float* ew1 = (const float*)d_in[2];
  const float* eb1 = (const float*)d_in[3];
  const float* ew2 = (const float*)d_in[4];
  const float* eb2 = (const float*)d_in[5];
  const float* nw1 = (const float*)d_in[6];
  const float* nb1 = (const float*)d_in[7];
  const float* nw2 = (const float*)d_in[8];
  const float* nb2 = (const float*)d_in[9];

  float* ws = (float*)d_ws;
  float* xbuf0 = ws;                    // 1024*128
  float* xbuf1 = ws + BN * DD;          // 1024*128
  float* xi_p  = ws + 2 * BN * DD;      // 1024*128
  float* xj_p  = ws + 3 * BN * DD;      // 1024*128  (2 MB total)

  const float* xin = x0;
  for (int l = 0; l < 3; ++l) {
    float* xo = (l == 2) ? (float*)d_out : ((l & 1) ? xbuf1 : xbuf0);
    in_proj_kernel<<<BN / 16, 256, 0, stream>>>(
        xin, ew1 + (size_t)l * 2 * DD * DD, eb1 + l * DD, xi_p, xj_p);
    in_fused_kernel<<<BN / 16, 256, 0, stream>>>(
        xin, adj, xi_p, xj_p, ew2 + (size_t)l * DD * DD, eb2 + l * DD,
        nw1 + (size_t)l * 2 * DD * DD, nb1 + l * DD,
        nw2 + (size_t)l * DD * DD, nb2 + l * DD, xo);
    xin = xo;
  }
}